// DecoderWithAttention_8615704396273
// MI455X (gfx1250) — compile-verified
//
#include <hip/hip_runtime.h>
#include <hip/hip_bf16.h>

// DecoderWithAttention for MI455X (gfx1250, wave32).
//   pre_e = enc @ Wa[:,H:].T + ba   precomputed once (64MB, L2-resident).
//   Per step: hW (WMMA) -> scores (tanh reduce) -> softmax+context ->
//             gx/gh (WMMA) -> GRU gates -> out (WMMA, written to d_out slice,
//             doubles as next step's x). 96 serial steps, graph-capturable.
// GEMM: fp32 WMMA 16x16x4; 4-wave blocks; B strip staged into LDS with
// double-buffered global_load_async_to_lds_b128 (ASYNCcnt) so the async
// engine overlaps chunk c+1 transfer with chunk c WMMAs; A fragments are
// register double-buffered.

#define Fdim 128
#define Hdim 512
#define Sdim 512
#define Bdim 64
#define Tdim 96
#define FH   (Fdim + Hdim)   // 640
#define G3   (3 * Hdim)      // 1536

#define KC     16            // k-chunk per LDS stage (4 wmma k-steps)
#define BPITCH 20            // LDS row pitch (dwords): 16B-aligned rows, no bank conflicts

typedef __attribute__((ext_vector_type(2))) float v2f;
typedef __attribute__((ext_vector_type(4))) float v4f;
typedef __attribute__((ext_vector_type(8))) float v8f;

__device__ __forceinline__ v8f wmma4(v2f a, v2f b, v8f c) {
    return __builtin_amdgcn_wmma_f32_16x16x4_f32(false, a, false, b, (short)0, c, false, false);
}

// ---------------------------------------------------------------------------
// C[m,n] = sum_k A[m,k]*W[n,k] (+bias[n]) (+C[m,n])
// Block: 128 threads = 4 waves; tile M=64 (wave w owns rows w*16..w*16+15),
// N=64. grid = (M/64, N/64, batch).
// V_WMMA_F32_16X16X4_F32 operand layout:
//   A 16x4: lane<16 -> M=lane, K=(k0,k0+1); lane>=16 -> M=lane-16, K=(k0+2,k0+3)
//   B 4x16 mirrors with N in place of M; B[k][n]=W[n][k] so both halves read a
//   contiguous float2 at W[n][k0+2*half].
//   C/D 16x16: VGPR r -> rows r (lanes 0-15) / r+8 (lanes 16-31).
// ---------------------------------------------------------------------------
__global__ __launch_bounds__(128) void gemm_wmma_f32(
    const float* __restrict__ A, int lda, long long strideAb,
    const float* __restrict__ W, int ldw,
    const float* __restrict__ bias,
    float* __restrict__ C, int ldc, long long strideCb,
    int K, int accumulate)
{
    __shared__ float smB[2 * 64 * BPITCH];     // double-buffered 64(n) x 16(k)

    const int tid  = threadIdx.x;
    const int wv   = tid >> 5;
    const int lane = tid & 31;
    const int half = lane >> 4;
    const int l16  = lane & 15;
    const int m0   = blockIdx.x * 64 + wv * 16;
    const int n0   = blockIdx.y * 64;

    const float* Ab = A + (long long)blockIdx.z * strideAb + (long long)(m0 + l16) * lda;
    float*       Cb = C + (long long)blockIdx.z * strideCb;

    // B staging: chunk = 64 rows x 16 k = 256 x 16B segments; each of the 128
    // threads moves two segments per chunk via async global->LDS b128.
    const int sA = tid, sB = tid + 128;
    const int nA = sA >> 2, pA = sA & 3;
    const int nB = sB >> 2, pB = sB & 3;
    const float* gA = W + (long long)(n0 + nA) * ldw + pA * 4;
    const float* gB = W + (long long)(n0 + nB) * ldw + pB * 4;
    const unsigned ldsA0 = (unsigned)(size_t)&smB[nA * BPITCH + pA * 4];
    const unsigned ldsB0 = (unsigned)(size_t)&smB[nB * BPITCH + pB * 4];

    auto stageB = [&](int k0, int buf) {
        unsigned la = ldsA0 + buf * (64 * BPITCH * 4);
        unsigned lb = ldsB0 + buf * (64 * BPITCH * 4);
        unsigned long long ga = (unsigned long long)(size_t)(gA + k0);
        unsigned long long gb = (unsigned long long)(size_t)(gB + k0);
        asm volatile("global_load_async_to_lds_b128 %0, %1, off"
                     :: "v"(la), "v"(ga) : "memory");
        asm volatile("global_load_async_to_lds_b128 %0, %1, off"
                     :: "v"(lb), "v"(gb) : "memory");
    };
    auto loadA = [&](int k0, v2f& x0, v2f& x1, v2f& x2, v2f& x3) {
        const float* p = Ab + k0 + 2 * half;
        x0 = *(const v2f*)(p);
        x1 = *(const v2f*)(p + 4);
        x2 = *(const v2f*)(p + 8);
        x3 = *(const v2f*)(p + 12);
    };

    v8f acc0 = {}, acc1 = {}, acc2 = {}, acc3 = {};
    v2f a0, a1, a2, a3, t0, t1, t2, t3;

    const int NC = K / KC;
    stageB(0, 0);
    loadA(0, a0, a1, a2, a3);
    asm volatile("s_wait_asynccnt 0x0" ::: "memory");
    __syncthreads();

    for (int c = 0; c < NC; ++c) {
        const int buf = c & 1;
        const bool more = (c + 1 < NC);
        if (more) {
            stageB((c + 1) * KC, buf ^ 1);       // overlaps with this chunk's WMMAs
            loadA((c + 1) * KC, t0, t1, t2, t3); // consumed next iteration
        }
        const float* lb = &smB[buf * 64 * BPITCH];
#pragma unroll
        for (int j = 0; j < 4; ++j) {
            const int kk = 4 * j + 2 * half;
            const v2f aa = (j == 0) ? a0 : (j == 1) ? a1 : (j == 2) ? a2 : a3;
            v2f b0 = *(const v2f*)&lb[(l16)      * BPITCH + kk];
            v2f b1 = *(const v2f*)&lb[(16 + l16) * BPITCH + kk];
            v2f b2 = *(const v2f*)&lb[(32 + l16) * BPITCH + kk];
            v2f b3 = *(const v2f*)&lb[(48 + l16) * BPITCH + kk];
            acc0 = wmma4(aa, b0, acc0);
            acc1 = wmma4(aa, b1, acc1);
            acc2 = wmma4(aa, b2, acc2);
            acc3 = wmma4(aa, b3, acc3);
        }
        asm volatile("s_wait_asynccnt 0x0" ::: "memory");
        __syncthreads();
        if (more) { a0 = t0; a1 = t1; a2 = t2; a3 = t3; }
    }

#pragma unroll
    for (int r = 0; r < 8; ++r) {
        const int m = m0 + r + 8 * half;
        const long long rowb = (long long)m * ldc;
        float vals[4] = {acc0[r], acc1[r], acc2[r], acc3[r]};
#pragma unroll
        for (int j = 0; j < 4; ++j) {
            const int n = n0 + 16 * j + l16;
            float v = vals[j];
            if (bias)       v += bias[n];
            if (accumulate) v += Cb[rowb + n];
            Cb[rowb + n] = v;
        }
    }
}

// ---------------------------------------------------------------------------
// scores[b,s] = bv + sum_h Wv[h] * tanh(pre_e[b,s,h] + hW[b,h]); wave per row.
// ---------------------------------------------------------------------------
__global__ __launch_bounds__(256) void attn_scores(
    const float* __restrict__ pre_e, const float* __restrict__ hW,
    const float* __restrict__ Wv, const float* __restrict__ bv,
    float* __restrict__ scores)
{
    const int row  = blockIdx.x * 8 + (threadIdx.x >> 5);  // b*S + s
    const int lane = threadIdx.x & 31;
    const int b    = row >> 9;

    const v4f* pe4 = (const v4f*)(pre_e + (long long)row * Hdim);
    const v4f* hw4 = (const v4f*)(hW + (long long)b * Hdim);
    const v4f* wv4 = (const v4f*)Wv;

    float acc = 0.f;
#pragma unroll
    for (int i = lane; i < Hdim / 4; i += 32) {
        v4f p = pe4[i], hh = hw4[i], wv = wv4[i];
        acc += wv.x * tanhf(p.x + hh.x);
        acc += wv.y * tanhf(p.y + hh.y);
        acc += wv.z * tanhf(p.z + hh.z);
        acc += wv.w * tanhf(p.w + hh.w);
    }
#pragma unroll
    for (int off = 16; off; off >>= 1) acc += __shfl_down(acc, off, 32);
    if (lane == 0) scores[row] = acc + bv[0];
}

// ---------------------------------------------------------------------------
// Per b: w = softmax(scores[b,:]); weighted[b,h] = sum_s w[s]*enc[b,s,h]
// ---------------------------------------------------------------------------
__global__ __launch_bounds__(256) void softmax_weighted(
    const float* __restrict__ scores, const float* __restrict__ enc,
    float* __restrict__ weighted)
{
    __shared__ float sw[Sdim];
    __shared__ float red[16];
    const int b   = blockIdx.x;
    const int tid = threadIdx.x;
    const int wid = tid >> 5;

    float s0 = scores[b * Sdim + tid];
    float s1 = scores[b * Sdim + tid + 256];

    float mx = fmaxf(s0, s1);
#pragma unroll
    for (int off = 16; off; off >>= 1) mx = fmaxf(mx, __shfl_down(mx, off, 32));
    if ((tid & 31) == 0) red[wid] = mx;
    __syncthreads();
    if (tid == 0) {
        float m = red[0];
#pragma unroll
        for (int i = 1; i < 8; ++i) m = fmaxf(m, red[i]);
        red[0] = m;
    }
    __syncthreads();
    mx = red[0];

    float e0 = __expf(s0 - mx), e1 = __expf(s1 - mx);
    sw[tid] = e0; sw[tid + 256] = e1;
    float sum = e0 + e1;
#pragma unroll
    for (int off = 16; off; off >>= 1) sum += __shfl_down(sum, off, 32);
    if ((tid & 31) == 0) red[8 + wid] = sum;
    __syncthreads();
    if (tid == 0) {
        float t = 0.f;
#pragma unroll
        for (int i = 0; i < 8; ++i) t += red[8 + i];
        red[8] = 1.0f / t;
    }
    __syncthreads();
    const float inv = red[8];
    sw[tid] *= inv; sw[tid + 256] *= inv;
    __syncthreads();

    const float* encb = enc + (long long)b * Sdim * Hdim;
    float a0 = 0.f, a1 = 0.f;
    for (int s = 0; s < Sdim; ++s) {
        const float w = sw[s];
        a0 += w * encb[(long long)s * Hdim + tid];
        a1 += w * encb[(long long)s * Hdim + tid + 256];
    }
    weighted[b * Hdim + tid]       = a0;
    weighted[b * Hdim + tid + 256] = a1;
}

// ---------------------------------------------------------------------------
// GRU gates (r,z,n order): h' = (1-z)*n + z*h
// ---------------------------------------------------------------------------
__global__ __launch_bounds__(256) void gru_gates(
    const float* __restrict__ gx, const float* __restrict__ gh,
    const float* __restrict__ h, float* __restrict__ hn)
{
    const int idx = blockIdx.x * 256 + threadIdx.x;   // < B*H
    const int b = idx >> 9;
    const int i = idx & (Hdim - 1);
    const float* gxb = gx + (long long)b * G3;
    const float* ghb = gh + (long long)b * G3;
    const float r = 1.f / (1.f + __expf(-(gxb[i] + ghb[i])));
    const float z = 1.f / (1.f + __expf(-(gxb[Hdim + i] + ghb[Hdim + i])));
    const float n = tanhf(gxb[2 * Hdim + i] + r * ghb[2 * Hdim + i]);
    hn[idx] = (1.f - z) * n + z * h[idx];
}

// ---------------------------------------------------------------------------
extern "C" void kernel_launch(void* const* d_in, const int* in_sizes, int n_in,
                              void* d_out, int out_size, void* d_ws, size_t ws_size,
                              hipStream_t stream) {
    const float* dec_input = (const float*)d_in[0];   // [B,F]
    const float* hidden    = (const float*)d_in[1];   // [1,B,H]
    const float* enc       = (const float*)d_in[2];   // [B,S,H]
    const float* Wa        = (const float*)d_in[5];   // [H,2H]
    const float* ba        = (const float*)d_in[6];   // [H]
    const float* Wv        = (const float*)d_in[7];   // [1,H]
    const float* bv        = (const float*)d_in[8];   // [1]
    const float* W_ih      = (const float*)d_in[9];   // [3H,F+H]
    const float* b_ih      = (const float*)d_in[10];  // [3H]
    const float* W_hh      = (const float*)d_in[11];  // [3H,H]
    const float* b_hh      = (const float*)d_in[12];  // [3H]
    const float* W_out     = (const float*)d_in[13];  // [F,H]
    const float* b_out     = (const float*)d_in[14];  // [F]
    float* out = (float*)d_out;                       // [B,T,F]

    // Workspace (floats), ~68.6 MB total; pre_e kept L2-resident (192MB L2).
    float* ws      = (float*)d_ws;
    float* pre_e   = ws;                                    // B*S*H
    float* hW      = pre_e + (long long)Bdim * Sdim * Hdim; // B*H
    float* sc      = hW    + Bdim * Hdim;                   // B*S
    float* weighted= sc    + Bdim * Sdim;                   // B*H
    float* hA      = weighted + Bdim * Hdim;                // B*H
    float* hB      = hA    + Bdim * Hdim;                   // B*H
    float* gx      = hB    + Bdim * Hdim;                   // B*3H
    float* gh      = gx    + (long long)Bdim * G3;          // B*3H

    // ---- Precompute pre_e = enc @ Wa[:,H:].T + ba  (batched over B) ----
    gemm_wmma_f32<<<dim3(Sdim / 64, Hdim / 64, Bdim), 128, 0, stream>>>(
        enc, Hdim, (long long)Sdim * Hdim,
        Wa + Hdim, 2 * Hdim, ba,
        pre_e, Hdim, (long long)Sdim * Hdim,
        Hdim, 0);

    const float* h_cur = hidden;  // [B,H]
    float* h_next = hA;

    for (int t = 0; t < Tdim; ++t) {
        const float* x   = (t == 0) ? dec_input : (out + (long long)(t - 1) * Fdim);
        const int    ldx = (t == 0) ? Fdim : Tdim * Fdim;

        // hW = h @ Wa[:,:H].T
        gemm_wmma_f32<<<dim3(1, Hdim / 64, 1), 128, 0, stream>>>(
            h_cur, Hdim, 0, Wa, 2 * Hdim, nullptr, hW, Hdim, 0, Hdim, 0);

        // scores = Wv . tanh(pre_e + hW) + bv
        attn_scores<<<dim3(Bdim * Sdim / 8), 256, 0, stream>>>(pre_e, hW, Wv, bv, sc);

        // softmax + weighted context
        softmax_weighted<<<dim3(Bdim), 256, 0, stream>>>(sc, enc, weighted);

        // gx = x @ W_ih[:,:F].T + b_ih ; gx += weighted @ W_ih[:,F:].T
        gemm_wmma_f32<<<dim3(1, G3 / 64, 1), 128, 0, stream>>>(
            x, ldx, 0, W_ih, FH, b_ih, gx, G3, 0, Fdim, 0);
        gemm_wmma_f32<<<dim3(1, G3 / 64, 1), 128, 0, stream>>>(
            weighted, Hdim, 0, W_ih + Fdim, FH, nullptr, gx, G3, 0, Hdim, 1);

        // gh = h @ W_hh.T + b_hh
        gemm_wmma_f32<<<dim3(1, G3 / 64, 1), 128, 0, stream>>>(
            h_cur, Hdim, 0, W_hh, Hdim, b_hh, gh, G3, 0, Hdim, 0);

        // GRU gates -> h_next
        gru_gates<<<dim3(Bdim * Hdim / 256), 256, 0, stream>>>(gx, gh, h_cur, h_next);

        // out_t = h_next @ W_out.T + b_out  -> [B,T,F] slice t
        gemm_wmma_f32<<<dim3(1, Fdim / 64, 1), 128, 0, stream>>>(
            h_next, Hdim, 0, W_out, Hdim, b_out, out + (long long)t * Fdim,
            Tdim * Fdim, 0, Hdim, 0);

        h_cur  = h_next;
        h_next = (h_cur == hA) ? hB : hA;
    }
}